// AllLoss_63170378989723
// MI455X (gfx1250) — compile-verified
//
#include <hip/hip_runtime.h>
#include <hip/hip_bf16.h>

typedef __attribute__((ext_vector_type(2))) float v2f;
typedef __attribute__((ext_vector_type(8))) float v8f;

#define NPTS   4096
#define NT     6          // 3 reflections + 3 rotations
#define NB     8
#define NTB    (NT*NB)    // 48
#define ROWS_PER_WG 512
#define NCHUNK (NPTS/ROWS_PER_WG)   // 8
#define BCHUNK 1024       // columns staged in LDS per pass
#define EPSF   1e-8f
#define INFBITS 0x7F800000u

// ---------------- workspace layout (floats) ----------------
#define WS_TRANS   0                               // NT*NB*NPTS*4   = 786432
#define WS_BTAB    (WS_TRANS + NT*NB*NPTS*4)       // NB*NPTS*4      = 131072
#define WS_COLMIN  (WS_BTAB + NB*NPTS*4)           // NT*NB*NPTS     = 196608 (u32 bits)
#define WS_ROWSUM  (WS_COLMIN + NT*NB*NPTS)        // NTB*NCHUNK     = 384
#define WS_REG     (WS_ROWSUM + NTB*NCHUNK)        // 1

__device__ __forceinline__ unsigned umin32(unsigned a, unsigned b) {
    return a < b ? a : b;
}

// =====================================================================
// Kernel 1: build transformed tables (x,y,z,aa) and B-table (-2x,-2y,-2z,bb);
//           init global col-min to +inf bits.
// =====================================================================
__global__ __launch_bounds__(256) void prep_kernel(
    const float* __restrict__ points,
    const float* __restrict__ plx, const float* __restrict__ ply, const float* __restrict__ plz,
    const float* __restrict__ rx,  const float* __restrict__ ry,  const float* __restrict__ rz,
    float* __restrict__ trans, float* __restrict__ btab, unsigned* __restrict__ colmin)
{
    int idx = blockIdx.x * 256 + threadIdx.x;        // (t*8+b)*4096 + n
    int n  = idx & (NPTS - 1);
    int tb = idx >> 12;
    int b  = tb & 7;
    int t  = tb >> 3;

    const float* p = points + ((size_t)b * NPTS + n) * 3;
    float px = p[0], py = p[1], pz = p[2];
    float ox, oy, oz;

    if (t < 3) {
        const float* pl = (t == 0 ? plx : (t == 1 ? ply : plz)) + b * 4;
        float nx = pl[0], ny = pl[1], nz = pl[2], d = pl[3];
        float denom = nx*nx + ny*ny + nz*nz + EPSF;
        float tt = (px*nx + py*ny + pz*nz + d) / denom;
        ox = px - 2.f*tt*nx; oy = py - 2.f*tt*ny; oz = pz - 2.f*tt*nz;
    } else {
        const float* q = (t == 3 ? rx : (t == 4 ? ry : rz)) + b * 4;
        float nrm = sqrtf(q[0]*q[0] + q[1]*q[1] + q[2]*q[2] + q[3]*q[3]) + EPSF;
        float qw = q[0]/nrm, qx = q[1]/nrm, qy = q[2]/nrm, qz = q[3]/nrm;
        float c1x = qy*pz - qz*py + qw*px;
        float c1y = qz*px - qx*pz + qw*py;
        float c1z = qx*py - qy*px + qw*pz;
        ox = px + 2.f*(qy*c1z - qz*c1y);
        oy = py + 2.f*(qz*c1x - qx*c1z);
        oz = pz + 2.f*(qx*c1y - qy*c1x);
    }
    float aa = ox*ox + oy*oy + oz*oz;
    float4* o = (float4*)trans + idx;
    *o = make_float4(ox, oy, oz, aa);
    colmin[idx] = INFBITS;
    if (t == 0) {
        float bb = px*px + py*py + pz*pz;
        ((float4*)btab)[b * NPTS + n] = make_float4(-2.f*px, -2.f*py, -2.f*pz, bb);
    }
}

// =====================================================================
// Kernel 2: WMMA chamfer. One wave = 64 rows (4 A-tiles), sweep 4096 cols.
// D tile = A(16x4) x B(4x16) with A row = (x,y,z,1), B col = (-2x',-2y',-2z',bb).
// Four independent accumulators -> 4 back-to-back WMMAs (no dest reuse, no
// copy-out movs), then a combined epilogue: pk_add, clamp, u32-bit mins.
// =====================================================================
__global__ __launch_bounds__(256) void chamfer_kernel(
    const float* __restrict__ trans, const float* __restrict__ btab,
    unsigned* __restrict__ colmin_g, float* __restrict__ rowsum_g)
{
    const int tb    = blockIdx.y;          // 0..47
    const int b     = tb & 7;
    const int chunk = blockIdx.x;          // 0..7
    const int wid   = threadIdx.x >> 5;
    const int lane  = threadIdx.x & 31;
    const int lh    = lane >> 4;           // K-half select
    const int ln    = lane & 15;           // row/col index within tile

    __shared__ float    s_bt[BCHUNK * 4];  // 16 KB staged B chunk
    __shared__ unsigned s_cm[NPTS];        // 16 KB per-WG col mins (bits)
    __shared__ float    s_ws[8];

    for (int i = threadIdx.x; i < NPTS; i += 256) s_cm[i] = INFBITS;

    const float* tr = trans + (size_t)tb * NPTS * 4;
    const int rowbase = chunk * ROWS_PER_WG + wid * 64;

    v2f      a_op[4];
    float    vaa[4][8];
    unsigned rmin[4][8];
#pragma unroll
    for (int mt = 0; mt < 4; ++mt) {
        int m0 = rowbase + mt * 16;
        const float* ap = tr + (m0 + ln) * 4 + lh * 2;
        a_op[mt].x = ap[0];                        // x (or z)
        a_op[mt].y = lh ? 1.0f : ap[1];            // y, or K=3 slot -> 1.0
#pragma unroll
        for (int r = 0; r < 8; ++r) {
            vaa[mt][r]  = tr[(m0 + lh * 8 + r) * 4 + 3];
            rmin[mt][r] = INFBITS;
        }
    }

    const float4* bt_g = (const float4*)btab + (size_t)b * NPTS;
    float4* s_bt4 = (float4*)s_bt;

    for (int cch = 0; cch < NPTS / BCHUNK; ++cch) {
        __syncthreads();
        for (int i = threadIdx.x; i < BCHUNK; i += 256)
            s_bt4[i] = bt_g[cch * BCHUNK + i];
        __syncthreads();

        for (int nt = 0; nt < BCHUNK / 16; ++nt) {
            const float* bp = s_bt + (nt * 16 + ln) * 4 + lh * 2;   // ds_load_b64
            v2f b_op; b_op.x = bp[0]; b_op.y = bp[1];

            // 4 independent WMMAs, back-to-back (no D->A/B hazards between them)
            v8f acc0 = {}, acc1 = {}, acc2 = {}, acc3 = {};
            acc0 = __builtin_amdgcn_wmma_f32_16x16x4_f32(
                    false, a_op[0], false, b_op, (short)0, acc0, false, false);
            acc1 = __builtin_amdgcn_wmma_f32_16x16x4_f32(
                    false, a_op[1], false, b_op, (short)0, acc1, false, false);
            acc2 = __builtin_amdgcn_wmma_f32_16x16x4_f32(
                    false, a_op[2], false, b_op, (short)0, acc2, false, false);
            acc3 = __builtin_amdgcn_wmma_f32_16x16x4_f32(
                    false, a_op[3], false, b_op, (short)0, acc3, false, false);

            unsigned cmin = INFBITS;
#pragma unroll
            for (int r = 0; r < 8; ++r) {
                float d0 = fmaxf(acc0[r] + vaa[0][r], 0.0f);
                float d1 = fmaxf(acc1[r] + vaa[1][r], 0.0f);
                float d2 = fmaxf(acc2[r] + vaa[2][r], 0.0f);
                float d3 = fmaxf(acc3[r] + vaa[3][r], 0.0f);
                unsigned b0 = __float_as_uint(d0);
                unsigned b1 = __float_as_uint(d1);
                unsigned b2 = __float_as_uint(d2);
                unsigned b3 = __float_as_uint(d3);
                rmin[0][r] = umin32(rmin[0][r], b0);
                rmin[1][r] = umin32(rmin[1][r], b1);
                rmin[2][r] = umin32(rmin[2][r], b2);
                rmin[3][r] = umin32(rmin[3][r], b3);
                cmin = umin32(cmin, umin32(umin32(b0, b1), umin32(b2, b3)));
            }
            atomicMin(&s_cm[cch * BCHUNK + nt * 16 + ln], cmin);
        }
    }

    // row-min: u32 min-tree across the 16 lanes of each half, then sum 64 rows/wave
    float lsum = 0.f;
#pragma unroll
    for (int mt = 0; mt < 4; ++mt)
#pragma unroll
        for (int r = 0; r < 8; ++r) {
            unsigned v = rmin[mt][r];
            v = umin32(v, (unsigned)__shfl_xor((int)v, 1, 32));
            v = umin32(v, (unsigned)__shfl_xor((int)v, 2, 32));
            v = umin32(v, (unsigned)__shfl_xor((int)v, 4, 32));
            v = umin32(v, (unsigned)__shfl_xor((int)v, 8, 32));
            lsum += __uint_as_float(v);        // replicated
        }
    float other = __shfl_xor(lsum, 16, 32);
    lsum = (lsum + other) * (1.0f / 16.0f);    // sum both halves (64 rows), then scale by 1/16
    if (lane == 0) s_ws[wid] = lsum;
    __syncthreads();
    if (threadIdx.x == 0) {
        float t = 0.f;
        for (int i = 0; i < 8; ++i) t += s_ws[i];
        rowsum_g[tb * NCHUNK + chunk] = t;
    }
    // flush col-mins
    for (int i = threadIdx.x; i < NPTS; i += 256)
        atomicMin(&colmin_g[(size_t)tb * NPTS + i], s_cm[i]);
}

// =====================================================================
// Kernel 3: regularizer (tiny)
// =====================================================================
__device__ float orth3(const float* a, const float* bvec, const float* c) {
    float v[3][3] = {{a[0],a[1],a[2]},{bvec[0],bvec[1],bvec[2]},{c[0],c[1],c[2]}};
    for (int i = 0; i < 3; ++i) {
        float nrm = sqrtf(v[i][0]*v[i][0] + v[i][1]*v[i][1] + v[i][2]*v[i][2]) + EPSF;
        v[i][0] /= nrm; v[i][1] /= nrm; v[i][2] /= nrm;
    }
    float s = 0.f;
    for (int i = 0; i < 3; ++i)
        for (int j = 0; j < 3; ++j) {
            float g = v[i][0]*v[j][0] + v[i][1]*v[j][1] + v[i][2]*v[j][2] - (i == j ? 1.f : 0.f);
            s += g * g;
        }
    return s;
}

__global__ void reg_kernel(const float* plx, const float* ply, const float* plz,
                           const float* rx, const float* ry, const float* rz,
                           float* regout)
{
    __shared__ float s[8];
    int b = threadIdx.x;
    if (b < 8)
        s[b] = orth3(plx + b*4, ply + b*4, plz + b*4)
             + orth3(rx + b*4 + 1, ry + b*4 + 1, rz + b*4 + 1);
    __syncthreads();
    if (threadIdx.x == 0) {
        float t = 0.f;
        for (int i = 0; i < 8; ++i) t += s[i];
        regout[0] = 25.0f * t / 8.0f;
    }
}

// =====================================================================
// Kernel 4: deterministic final reduction
// =====================================================================
__global__ __launch_bounds__(256) void reduce_kernel(
    const unsigned* __restrict__ colmin, const float* __restrict__ rowsum,
    const float* __restrict__ regv, float* __restrict__ out)
{
    __shared__ float s[256];
    float acc = 0.f;
    for (int i = threadIdx.x; i < NT*NB*NPTS; i += 256) acc += __uint_as_float(colmin[i]);
    for (int i = threadIdx.x; i < NTB*NCHUNK;  i += 256) acc += rowsum[i];
    s[threadIdx.x] = acc;
    __syncthreads();
    for (int st = 128; st > 0; st >>= 1) {
        if (threadIdx.x < st) s[threadIdx.x] += s[threadIdx.x + st];
        __syncthreads();
    }
    if (threadIdx.x == 0)
        out[0] = regv[0] + s[0] * (1.0f / (float)(NB * NPTS));
}

// =====================================================================
extern "C" void kernel_launch(void* const* d_in, const int* in_sizes, int n_in,
                              void* d_out, int out_size, void* d_ws, size_t ws_size,
                              hipStream_t stream)
{
    (void)in_sizes; (void)n_in; (void)out_size; (void)ws_size;
    const float* points = (const float*)d_in[0];
    // d_in[1] = near_point (unused by reference)
    const float* plx = (const float*)d_in[2];
    const float* ply = (const float*)d_in[3];
    const float* plz = (const float*)d_in[4];
    const float* rx  = (const float*)d_in[5];
    const float* ry  = (const float*)d_in[6];
    const float* rz  = (const float*)d_in[7];

    float*    ws     = (float*)d_ws;
    float*    trans  = ws + WS_TRANS;
    float*    btab   = ws + WS_BTAB;
    unsigned* colmin = (unsigned*)(ws + WS_COLMIN);
    float*    rowsum = ws + WS_ROWSUM;
    float*    regv   = ws + WS_REG;

    prep_kernel<<<(NT*NB*NPTS)/256, 256, 0, stream>>>(points, plx, ply, plz, rx, ry, rz,
                                                      trans, btab, colmin);
    reg_kernel<<<1, 32, 0, stream>>>(plx, ply, plz, rx, ry, rz, regv);
    chamfer_kernel<<<dim3(NCHUNK, NTB), 256, 0, stream>>>(trans, btab, colmin, rowsum);
    reduce_kernel<<<1, 256, 0, stream>>>(colmin, rowsum, regv, (float*)d_out);
}